// FixedDithering_79285096284312
// MI455X (gfx1250) — compile-verified
//
#include <hip/hip_runtime.h>
#include <stdint.h>

#define IMG_W 512
#define IMG_R 512
#define CHUNK 128            // columns per TDM chunk
#define ROWS  32             // rows per row-block == wave32 lanes
#define NRB   16             // row-blocks per image (512/32)
#define NCHUNK_IMG 64        // chunks per image (16 * 4)
#define TSTEPS (IMG_W + 2 * (ROWS - 1))   // 574 wavefront steps per row-block

#if defined(__gfx1250__)

typedef __attribute__((ext_vector_type(4))) unsigned int tdm_v4u;
typedef __attribute__((ext_vector_type(8))) int          tdm_v8i;
typedef __attribute__((ext_vector_type(4))) int          tdm_v4i;

#if __has_builtin(__builtin_amdgcn_tensor_load_to_lds)
#define HAVE_TDM_BUILTIN 1
#else
#define HAVE_TDM_BUILTIN 0
#endif

#if __has_builtin(__builtin_amdgcn_s_wait_tensorcnt)
#define WAIT_TCNT(n) __builtin_amdgcn_s_wait_tensorcnt(n)
#else
#define WAIT_TCNT(n) asm volatile("s_wait_tensorcnt %0" :: "i"(n) : "memory")
#endif
#define WAIT_DSCNT0() asm volatile("s_wait_dscnt 0x0" ::: "memory")

__device__ __forceinline__ unsigned lds_addr32(const void* p) {
  // low 32 bits of the flat address are the LDS byte offset
  return (unsigned)(unsigned long long)(uintptr_t)p;
}

__device__ __forceinline__ float clamp01(float v) {
#if __has_builtin(__builtin_amdgcn_fmed3f)
  return __builtin_amdgcn_fmed3f(v, 0.0f, 1.0f);   // v_med3_f32: 1-op clamp
#else
  return fminf(fmaxf(v, 0.0f), 1.0f);
#endif
}

// Shift err from lane L-1 to lane L on the VALU (DPP row_shr:1), patching the
// 16-lane row boundary with v_readlane.  Falls back to __shfl_up (ds_bpermute).
__device__ __forceinline__ float shift_up1(float v, bool is_lane16) {
#if __has_builtin(__builtin_amdgcn_update_dpp) && __has_builtin(__builtin_amdgcn_readlane)
  int vi = __builtin_bit_cast(int, v);
  int sh = __builtin_amdgcn_update_dpp(0, vi, 0x111 /*row_shr:1*/, 0xf, 0xf, true);
  int l15 = __builtin_amdgcn_readlane(vi, 15);
  float r = __builtin_bit_cast(float, sh);
  float f15 = __builtin_bit_cast(float, l15);
  return is_lane16 ? f15 : r;
#else
  (void)is_lane16;
  return __shfl_up(v, 1);
#endif
}

// Issue one TDM transfer of a 32x128 f32 tile (row stride 512 elements in
// global memory, rows packed contiguously in LDS).  D# built per
// CDNA5 ISA.
__device__ __forceinline__ void tdm_xfer(int is_store, unsigned lds, const void* gptr) {
  unsigned long long ga = (unsigned long long)(uintptr_t)gptr;
  tdm_v4u g0;
  g0.x = 1u;                                    // count=1 valid descriptor
  g0.y = lds;                                   // lds_addr (bytes)
  g0.z = (unsigned)ga;                          // global_addr[31:0]
  g0.w = ((unsigned)(ga >> 32) & 0x01FFFFFFu)   // global_addr[56:32]
         | (2u << 30);                          // type = 2 ("image")
  tdm_v8i g1;
  g1[0] = (int)(2u << 16);                  // data_size=2 (4B); no multicast/pad
  g1[1] = (int)((unsigned)CHUNK << 16);     // tensor_dim0[15:0] = 128 (bits 63:48)
  g1[2] = (int)((unsigned)ROWS  << 16);     // tensor_dim0 hi = 0 | tensor_dim1 lo = 32
  g1[3] = (int)((unsigned)CHUNK << 16);     // tensor_dim1 hi = 0 | tile_dim0 = 128
  g1[4] = ROWS;                             // tile_dim1 = 32 | tile_dim2 = 0
  g1[5] = IMG_W;                            // tensor_dim0_stride lo32 = 512 elements
  g1[6] = 0;                                // stride0 hi | stride1 lo
  g1[7] = 0;                                // stride1 hi
  tdm_v4i z4 = {0, 0, 0, 0};
#if HAVE_TDM_BUILTIN
#if __clang_major__ >= 23
  tdm_v8i z8 = {0, 0, 0, 0, 0, 0, 0, 0};
  if (is_store) __builtin_amdgcn_tensor_store_from_lds(g0, g1, z4, z4, z8, 0);
  else          __builtin_amdgcn_tensor_load_to_lds  (g0, g1, z4, z4, z8, 0);
#else
  if (is_store) __builtin_amdgcn_tensor_store_from_lds(g0, g1, z4, z4, 0);
  else          __builtin_amdgcn_tensor_load_to_lds  (g0, g1, z4, z4, 0);
#endif
#else
  if (is_store) asm volatile("tensor_store_from_lds %0, %1" :: "s"(g0), "s"(g1) : "memory");
  else          asm volatile("tensor_load_to_lds %0, %1"    :: "s"(g0), "s"(g1) : "memory");
#endif
}
#endif // __gfx1250__

// One wave32 per image.  Lane L owns row r0+L of the current 32-row block and
// runs 2 columns behind lane L-1 (anti-diagonal wavefront).  Up-row errors
// arrive via DPP lane-shift; lane 0 takes them from the LDS buffer written by
// lane 31 of the previous row-block.  Input tiles stream in / output tiles
// stream out via the Tensor Data Mover through a 3-slot LDS ring (software
// pipeline tracked with TENSORcnt; LDS->TDM ordering via s_wait_dscnt).
__global__ __launch_bounds__(32) void fs_dither_kernel(const float* __restrict__ x,
                                                       float* __restrict__ y) {
#if defined(__gfx1250__)
  __shared__ float ring[3][ROWS * CHUNK];   // 48 KB: 3 x (32x128) f32 tiles
  __shared__ float perr[640];               // prev-row errors, index = 1+col, zero-padded

  const int    lane = (int)threadIdx.x;
  const bool   is0  = (lane == 0);
  const bool   is16 = (lane == 16);
  const bool   is31 = (lane == ROWS - 1);
  const size_t img  = blockIdx.x;
  const float* xin = x + img * (size_t)(IMG_R * IMG_W);
  float*       yo  = y + img * (size_t)(IMG_R * IMG_W);

  for (int i = lane; i < 640; i += 32) perr[i] = 0.0f;   // row -1 has zero errors
  __syncthreads();

  // Prologue: fill the ring with chunks 0,1,2 of row-block 0.
  tdm_xfer(0, lds_addr32(&ring[0][0]), xin + 0 * CHUNK);
  tdm_xfer(0, lds_addr32(&ring[1][0]), xin + 1 * CHUNK);
  tdm_xfer(0, lds_addr32(&ring[2][0]), xin + 2 * CHUNK);
  int g_next = 3;   // next global chunk index to prefetch (0..63)

  const float WUL = 1.0f / 16.0f, WU = 5.0f / 16.0f;
  const float WUR = 3.0f / 16.0f, WLf = 7.0f / 16.0f;

  for (int rb = 0; rb < NRB; ++rb) {
    const int gbase = rb * 4;
    WAIT_TCNT(2);   // chunk gbase resident (trailing store + lead prefetch in flight)

    float eL = 0.0f;                          // left-neighbor error (row-local)
    float u_m1 = 0.0f, u0 = 0.0f, u1 = 0.0f;  // up-row errors at cols c-1,c,c+1
    if (is0) { u0 = perr[1]; u1 = perr[2]; }

    // One wavefront step: lane L handles column c = t - 2L of its row.
    auto step = [&](int t) {
      const int  c      = t - 2 * lane;
      const bool active = (unsigned)c < (unsigned)IMG_W;
      float err = 0.0f;
      if (active) {
        const int   slot = (gbase + (c >> 7)) % 3;
        const int   idx  = (lane << 7) + (c & (CHUNK - 1));
        const float xv   = ring[slot][idx];
        // x01 = clip((clip(x,-1,1)+1)/2) == clamp01(fma(x,.5,.5))
        const float x01 = clamp01(fmaf(xv, 0.5f, 0.5f));
        const float up  = fmaf(WUL, u_m1, fmaf(WU, u0, WUR * u1));
        const float v   = clamp01(fmaf(WLf, eL, x01 + up));   // v_med3 clamp
        const float q   = __builtin_rintf(v);   // round-half-even (v_rndne_f32)
        err = v - q;
        ring[slot][idx] = fmaf(q, 2.0f, -1.0f); // write +/-1 output in place
        if (is31) perr[1 + c] = err;            // feed next row-block
      }
      eL = err;
      const float eup = shift_up1(err, is16);   // error from the row above
      const float nl  = is0 ? perr[t + 3] : eup;
      u_m1 = u0; u0 = u1; u1 = nl;
    };

    // Store fully-consumed chunk k, prefetch chunk k+3 into the freed slot.
    auto consume = [&](int k) {
      const int g = gbase + k;
      WAIT_DSCNT0();   // wave's LDS output writes visible to the TDM store
      tdm_xfer(1, lds_addr32(&ring[g % 3][0]),
               yo + (size_t)(rb * ROWS) * IMG_W + k * CHUNK);
      if (g_next < NCHUNK_IMG) {
        const int rb2 = g_next >> 2, k2 = g_next & 3;
        tdm_xfer(0, lds_addr32(&ring[g_next % 3][0]),
                 xin + (size_t)(rb2 * ROWS) * IMG_W + k2 * CHUNK);
        ++g_next;
      }
    };

    // Segmented wavefront: waits / TDM issues only at chunk boundaries.
    for (int t = 0;   t < 128; ++t) step(t);     // chunk 0 only
    if (rb == 0) WAIT_TCNT(1); else WAIT_TCNT(2);  // chunk 1 resident
    for (int t = 128; t < 190; ++t) step(t);     // chunks 0+1
    consume(0);                                  // chunk 0 done after t=189
    for (int t = 190; t < 256; ++t) step(t);     // chunk 1
    WAIT_TCNT(2);                                // chunk 2 resident
    for (int t = 256; t < 318; ++t) step(t);     // chunks 1+2
    consume(1);
    for (int t = 318; t < 384; ++t) step(t);     // chunk 2
    WAIT_TCNT(2);                                // chunk 3 resident
    for (int t = 384; t < 446; ++t) step(t);     // chunks 2+3
    consume(2);
    for (int t = 446; t < TSTEPS; ++t) step(t);  // chunk 3 tail
    consume(3);                                  // last chunk of the row-block
  }
  WAIT_TCNT(0);   // drain all TDM stores before wave end
#endif
}

extern "C" void kernel_launch(void* const* d_in, const int* in_sizes, int n_in,
                              void* d_out, int out_size, void* d_ws, size_t ws_size,
                              hipStream_t stream) {
  (void)n_in; (void)out_size; (void)d_ws; (void)ws_size;
  const float* x = (const float*)d_in[0];
  float*       y = (float*)d_out;
  const int n_img = in_sizes[0] / (IMG_R * IMG_W);   // 32*3 = 96 images
  fs_dither_kernel<<<dim3((unsigned)n_img), dim3(32), 0, stream>>>(x, y);
}